// SingleStageDetector_11338713661593
// MI455X (gfx1250) — compile-verified
//
#include <hip/hip_runtime.h>
#include <hip/hip_bf16.h>

// Problem constants (from reference): B=16, A=9, H=64, W=64, N=40
#define B_  16
#define A_  9
#define H_  64
#define W_  64
#define N_  40
#define HW_ (H_ * W_)          // 4096
#define P_  (A_ * HW_)         // 36864 proposals per batch
#define TP_ (B_ * P_)          // 589824 total proposals
#define PROPS_PER_BLOCK 64     // tile: 64 proposals x 40 boxes = 2560 outputs
#define THREADS 256            // 8 wave32 waves

#if defined(__HIP_DEVICE_COMPILE__) && __has_builtin(__builtin_amdgcn_global_load_async_to_lds_b128)
#define HAVE_ASYNC_LDS 1
#else
#define HAVE_ASYNC_LDS 0
#endif

#if HAVE_ASYNC_LDS
// Builtin signature (from hipcc diagnostic): param 0 is
// 'int __attribute__((vector_size(16))) addrspace-global *'
typedef int v4i_ __attribute__((vector_size(16)));
typedef __attribute__((address_space(1))) v4i_ g_v4i;   // global int4
typedef __attribute__((address_space(3))) v4i_ l_v4i;   // LDS int4
#endif

static __device__ __forceinline__ void wait_async_zero() {
#if defined(__HIP_DEVICE_COMPILE__)
#if __has_builtin(__builtin_amdgcn_s_wait_asynccnt)
    __builtin_amdgcn_s_wait_asynccnt(0);
#else
    asm volatile("s_wait_asynccnt 0x0" ::: "memory");
#endif
#endif
}

// One block = 64 consecutive proposals (same batch b, same anchor a) x all 40 boxes.
// Stage bboxes via CDNA5 async global->LDS; decode proposals once into LDS
// (so the 2x expf per proposal is NOT redone 40x); then 2560 coalesced stores.
__global__ void __launch_bounds__(THREADS)
detector_iou_kernel(const float* __restrict__ anc,      // (9,2)
                    const float2* __restrict__ grid2,   // (B,H,W,2) as float2
                    const float4* __restrict__ off4,    // (B,A,H,W,4) as float4
                    const float* __restrict__ bboxes,   // (B,N,5)
                    float* __restrict__ out)            // (B, P_, N_)
{
    __shared__ float4 sprop[PROPS_PER_BLOCK];       // proposal boxes
    __shared__ float  sareap[PROPS_PER_BLOCK];      // proposal areas
    __shared__ float  sbx1[N_], sby1[N_], sbx2[N_], sby2[N_], sareab[N_];
    __shared__ __align__(16) float sraw[5 * N_];    // 200 floats = 800 B staged bboxes

    const int tid = threadIdx.x;
    const int p0  = blockIdx.x * PROPS_PER_BLOCK;   // first global proposal of tile
    const int b   = p0 / P_;                        // constant per block (P_ % 64 == 0)

    // ---- stage this batch's bboxes into LDS (async DMA path on gfx1250) ----
#if HAVE_ASYNC_LDS
    if (tid < (5 * N_) / 4) {                       // 50 lanes x b128 = 800 bytes
        __builtin_amdgcn_global_load_async_to_lds_b128(
            (g_v4i*)(bboxes + (size_t)b * 5 * N_ + tid * 4),
            (l_v4i*)(&sraw[tid * 4]),
            0, 0);
    }
#else
    if (tid < 5 * N_) sraw[tid] = bboxes[(size_t)b * 5 * N_ + tid];
#endif

    // ---- decode 64 proposals (one thread each; exp computed exactly once) ----
    if (tid < PROPS_PER_BLOCK) {
        const int r  = (p0 + tid) - b * P_;         // proposal index within batch
        const int a  = r / HW_;                     // constant per block too
        const int hw = r - a * HW_;

        const float2 g   = grid2[b * HW_ + hw];
        const float  hfw = anc[a * 2 + 0] * 0.5f;
        const float  hfh = anc[a * 2 + 1] * 0.5f;
        const float4 o   = off4[(size_t)(b * A_ + a) * HW_ + hw];

        // mirror reference op order for bit-closeness
        const float a0 = g.x - hfw, a1 = g.y - hfh;
        const float a2 = g.x + hfw, a3 = g.y + hfh;
        const float cx = (a0 + a2) * 0.5f + o.x;
        const float cy = (a1 + a3) * 0.5f + o.y;
        const float pw = (a2 - a0) * expf(o.z);
        const float ph = (a3 - a1) * expf(o.w);

        const float x1 = cx - pw * 0.5f, y1 = cy - ph * 0.5f;
        const float x2 = cx + pw * 0.5f, y2 = cy + ph * 0.5f;
        sprop[tid]  = make_float4(x1, y1, x2, y2);
        sareap[tid] = (x2 - x1) * (y2 - y1);        // reference computes area from coords
    }

#if HAVE_ASYNC_LDS
    wait_async_zero();
#endif
    __syncthreads();

    if (tid < N_) {
        const float x1 = sraw[tid * 5 + 0], y1 = sraw[tid * 5 + 1];
        const float x2 = sraw[tid * 5 + 2], y2 = sraw[tid * 5 + 3];
        sbx1[tid] = x1; sby1[tid] = y1; sbx2[tid] = x2; sby2[tid] = y2;
        sareab[tid] = (x2 - x1) * (y2 - y1);
    }
    __syncthreads();

    // ---- 64 x 40 = 2560 IoU outputs, coalesced b32 stores ----
    const size_t base = (size_t)p0 * N_;
    #pragma unroll
    for (int i = tid; i < PROPS_PER_BLOCK * N_; i += THREADS) {
        const int pl = i / N_;
        const int n  = i - pl * N_;
        const float4 pr = sprop[pl];

        const float xtl = fmaxf(pr.x, sbx1[n]);
        const float ytl = fmaxf(pr.y, sby1[n]);
        const float xbr = fminf(pr.z, sbx2[n]);
        const float ybr = fminf(pr.w, sby2[n]);
        const float inter = fmaxf(xbr - xtl, 0.0f) * fmaxf(ybr - ytl, 0.0f);

        out[base + i] = inter / (sareap[pl] + sareab[n] - inter);
    }
}

extern "C" void kernel_launch(void* const* d_in, const int* in_sizes, int n_in,
                              void* d_out, int out_size, void* d_ws, size_t ws_size,
                              hipStream_t stream) {
    (void)in_sizes; (void)n_in; (void)d_ws; (void)ws_size; (void)out_size;
    const float* anc     = (const float*)d_in[0];   // (9,2)
    const float* grid    = (const float*)d_in[1];   // (16,64,64,2)
    const float* offsets = (const float*)d_in[2];   // (16,9,64,64,4)
    const float* bboxes  = (const float*)d_in[3];   // (16,40,5)
    float* out = (float*)d_out;                     // (16, 36864, 40) fp32

    const int blocks = TP_ / PROPS_PER_BLOCK;       // 9216
    detector_iou_kernel<<<blocks, THREADS, 0, stream>>>(
        anc, (const float2*)grid, (const float4*)offsets, bboxes, out);
}